// GINNet_9251359555630
// MI455X (gfx1250) — compile-verified
//
#include <hip/hip_runtime.h>

typedef __attribute__((ext_vector_type(2))) float v2f;
typedef __attribute__((ext_vector_type(8))) float v8f;

#define N_NODES    10000
#define N_EDGES    160000
#define N_GRAPHS   64
#define FIRST_DIM  128
#define EMB        300
#define NUM_CLASSES 10
#define BN_EPS     1e-5f

// ---------------------------------------------------------------- utilities

__global__ void k_zero(float* __restrict__ p, long n) {
  long stride = (long)gridDim.x * blockDim.x;
  for (long i = blockIdx.x * (long)blockDim.x + threadIdx.x; i < n; i += stride)
    p[i] = 0.f;
}

__global__ void k_broadcast_row(float* __restrict__ out, const float* __restrict__ row,
                                int G, int D) {
  long n = (long)G * D;
  long stride = (long)gridDim.x * blockDim.x;
  for (long i = blockIdx.x * (long)blockDim.x + threadIdx.x; i < n; i += stride)
    out[i] = row[i % D];
}

__global__ void k_counts(const int* __restrict__ batch, float* __restrict__ counts, int Nn) {
  int i = blockIdx.x * blockDim.x + threadIdx.x;
  if (i < Nn) atomicAdd(&counts[batch[i]], 1.f);
}

// h[n][:] += vnf[batch[n]][:]
__global__ void k_add_vnf(float* __restrict__ h, const float* __restrict__ vnf,
                          const int* __restrict__ batch, int D) {
  int n = blockIdx.x;
  int g = batch[n];
  for (int d = threadIdx.x; d < D; d += blockDim.x)
    h[(long)n * D + d] += vnf[(long)g * D + d];
}

// t[dst[e]][:] += h[src[e]][:]   (t pre-initialized with h for the self term)
__global__ void k_scatter_edges(const float* __restrict__ h, const int* __restrict__ esrc,
                                const int* __restrict__ edst, float* __restrict__ t, int D) {
  int e = blockIdx.x;
  int s = esrc[e];
  int d2 = edst[e];
  for (int d = threadIdx.x; d < D; d += blockDim.x)
    atomicAdd(&t[(long)d2 * D + d], h[(long)s * D + d]);
}

// pooled[batch[n]][:] += h[n][:]
__global__ void k_pool_add(const float* __restrict__ h, const int* __restrict__ batch,
                           float* __restrict__ pooled, int D) {
  int n = blockIdx.x;
  int g = batch[n];
  for (int d = threadIdx.x; d < D; d += blockDim.x)
    atomicAdd(&pooled[(long)g * D + d], h[(long)n * D + d]);
}

// partial column sums / sums of squares; sums[0..N-1]=sum, sums[N..2N-1]=sumsq
__global__ void k_colstats(const float* __restrict__ X, float* __restrict__ sums,
                           int M, int N, int rowsPerBlock) {
  int r0 = blockIdx.x * rowsPerBlock;
  int r1 = min(M, r0 + rowsPerBlock);
  for (int c = threadIdx.x; c < N; c += blockDim.x) {
    float s = 0.f, ss = 0.f;
    for (int r = r0; r < r1; ++r) {
      float v = X[(long)r * N + c];
      s += v;
      ss += v * v;
    }
    atomicAdd(&sums[c], s);
    atomicAdd(&sums[N + c], ss);
  }
}

// in-place training-mode BN (+ optional ReLU), population variance like jnp.var
__global__ void k_bn_apply(float* __restrict__ X, const float* __restrict__ sums,
                           const float* __restrict__ g, const float* __restrict__ b,
                           int M, int N, int doRelu) {
  long n = (long)M * N;
  long stride = (long)gridDim.x * blockDim.x;
  float invM = 1.f / (float)M;
  for (long i = blockIdx.x * (long)blockDim.x + threadIdx.x; i < n; i += stride) {
    int c = (int)(i % N);
    float mean = sums[c] * invM;
    float var  = sums[N + c] * invM - mean * mean;
    float y = (X[i] - mean) * rsqrtf(var + BN_EPS) * g[c] + b[c];
    if (doRelu) y = fmaxf(y, 0.f);
    X[i] = y;
  }
}

__global__ void k_mean_div(float* __restrict__ pooled, const float* __restrict__ counts,
                           int G, int D) {
  long n = (long)G * D;
  long stride = (long)gridDim.x * blockDim.x;
  for (long i = blockIdx.x * (long)blockDim.x + threadIdx.x; i < n; i += stride)
    pooled[i] /= fmaxf(counts[i / D], 1.f);
}

// Repack W[K][N] -> Wp[k/4][N][4] so a lane's B fragment (rows kk,kk+1 at one
// column) is a single aligned b64 load.
__global__ void k_pack_w(const float* __restrict__ W, float* __restrict__ Wp,
                         int N, int K) {
  long n = (long)K * N;
  long stride = (long)gridDim.x * blockDim.x;
  for (long i = blockIdx.x * (long)blockDim.x + threadIdx.x; i < n; i += stride) {
    int k = (int)(i / N), c = (int)(i % N);
    Wp[(long)(k >> 2) * (4 * N) + c * 4 + (k & 3)] = W[i];
  }
}

// ------------------------------------------------------------ fp32 WMMA GEMM
// C[M,N] = A[M,K] @ W[K,N] + bias[N], W pre-packed by k_pack_w.
// One wave computes a 16x64 C tile (4 accumulators) with V_WMMA_F32_16X16X4_F32.
// N,K are compile-time: loop addressing is loop-invariant per-lane offsets plus
// a base advanced by an immediate (no 64-bit multiplies in the hot loop).
// Out-of-range rows/cols handled by CLAMPING load indices (garbage only lands
// in never-stored C elements) so EXEC stays all-1s for WMMA.
template<int N, int K>
__global__ void k_gemm_wmma_f32(const float* __restrict__ A, const float* __restrict__ Wp,
                                const float* __restrict__ bias, float* __restrict__ C,
                                int M) {
  const int lane = threadIdx.x & 31;
  const int wib  = threadIdx.x >> 5;
  const int gw   = blockIdx.x * (blockDim.x >> 5) + wib;
  const int mT = (M + 15) >> 4;
  constexpr int nT = (N + 63) >> 6;
  if (gw >= mT * nT) return;            // wave-uniform exit
  const int mt = gw % mT;               // adjacent waves share the same W tile
  const int nt = gw / mT;
  const int m0 = mt << 4, n0 = nt << 6;

  const int half = lane >> 4;           // 0: lanes 0-15, 1: lanes 16-31
  const int l    = lane & 15;

  int arow = m0 + l;  if (arow >= M) arow = M - 1;     // clamp (never stored)
  const float* apt = A + (long)arow * K + 2 * half;    // 8B aligned (K even, all uses)
  int boff[4];                                         // loop-invariant lane offsets
  #pragma unroll
  for (int j = 0; j < 4; ++j) {
    int c = n0 + 16 * j + l;
    if (c >= N) c = N - 1;                             // clamp (never stored)
    boff[j] = c * 4 + 2 * half;
  }

  v8f acc[4];
  #pragma unroll
  for (int j = 0; j < 4; ++j) acc[j] = (v8f){0.f,0.f,0.f,0.f,0.f,0.f,0.f,0.f};

  #pragma unroll 2
  for (int k4 = 0; k4 < K / 4; ++k4) {
    // A 16x4 frag: lanes0-15 hold K=k,k+1; lanes16-31 hold K=k+2,k+3 (b64 load)
    v2f a = *(const v2f*)(apt + 4 * k4);
    const float* wq = Wp + (long)k4 * (4 * N);
    v2f bf[4];
    #pragma unroll
    for (int j = 0; j < 4; ++j)
      bf[j] = *(const v2f*)(wq + boff[j]);             // rows kk,kk+1: one b64
    #pragma unroll
    for (int j = 0; j < 4; ++j)
      acc[j] = __builtin_amdgcn_wmma_f32_16x16x4_f32(false, a, false, bf[j],
                                                     (short)0, acc[j], false, false);
  }

  // C/D layout: VGPR r -> row m0 + r + 8*half, col = subtile base + (lane&15)
  const int rb = m0 + half * 8;
  #pragma unroll
  for (int j = 0; j < 4; ++j) {
    int c = n0 + 16 * j + l;
    if (c < N) {
      float bv = bias[c];
      float* cp = C + (long)rb * N + c;
      #pragma unroll
      for (int r = 0; r < 8; ++r)
        if (rb + r < M) cp[(long)r * N] = acc[j][r] + bv;
    }
  }
}

// ------------------------------------------------------------------- driver

template<int N, int K>
static void launch_gemm(const float* A, const float* W, const float* bias, float* C,
                        int M, float* wpack, hipStream_t stream) {
  long nw = (long)K * N;
  k_pack_w<<<(int)((nw + 255) / 256), 256, 0, stream>>>(W, wpack, N, K);
  int waves  = ((M + 15) / 16) * ((N + 63) / 64);
  int blocks = (waves + 7) / 8;                        // 8 waves (256 thr) / block
  k_gemm_wmma_f32<N, K><<<blocks, 256, 0, stream>>>(A, wpack, bias, C, M);
}

extern "C" void kernel_launch(void* const* d_in, const int* in_sizes, int n_in,
                              void* d_out, int out_size, void* d_ws, size_t ws_size,
                              hipStream_t stream) {
  (void)in_sizes; (void)n_in; (void)out_size; (void)ws_size;

  const float* x     = (const float*)d_in[0];
  const int*   eidx  = (const int*)d_in[1];
  const int*   batch = (const int*)d_in[2];
  const float* ve    = (const float*)d_in[3];
  const float* c1_w1 = (const float*)d_in[4];
  const float* c1_b1 = (const float*)d_in[5];
  const float* c1_g  = (const float*)d_in[6];
  const float* c1_bt = (const float*)d_in[7];
  const float* c1_w2 = (const float*)d_in[8];
  const float* c1_b2 = (const float*)d_in[9];
  const float* cw1   = (const float*)d_in[10];
  const float* cb1   = (const float*)d_in[11];
  const float* cg    = (const float*)d_in[12];
  const float* cbt   = (const float*)d_in[13];
  const float* cw2   = (const float*)d_in[14];
  const float* cb2   = (const float*)d_in[15];
  const float* bn0_g = (const float*)d_in[16];
  const float* bn0_b = (const float*)d_in[17];
  const float* bns_g = (const float*)d_in[18];
  const float* bns_b = (const float*)d_in[19];
  const float* vw1   = (const float*)d_in[20];
  const float* vb1   = (const float*)d_in[21];
  const float* vg1   = (const float*)d_in[22];
  const float* vbt1  = (const float*)d_in[23];
  const float* vw2   = (const float*)d_in[24];
  const float* vb2   = (const float*)d_in[25];
  const float* vg2   = (const float*)d_in[26];
  const float* vbt2  = (const float*)d_in[27];
  const float* pw    = (const float*)d_in[28];
  const float* pb    = (const float*)d_in[29];
  float* out = (float*)d_out;

  const int* esrc = eidx;
  const int* edst = eidx + N_EDGES;

  // workspace layout (floats); total ~12.5M floats (~50 MB)
  float* ws     = (float*)d_ws;
  float* h      = ws;                                   // 10000*300
  float* t      = h + (long)N_NODES * EMB;              // 10000*300 (conv1 uses *128)
  float* u      = t + (long)N_NODES * EMB;              // 10000*600
  float* vnf    = u + (long)N_NODES * 2 * EMB;          // 64*300
  float* pooled = vnf + (long)N_GRAPHS * EMB;           // 64*300
  float* vz     = pooled + (long)N_GRAPHS * EMB;        // 64*600
  float* sums   = vz + (long)N_GRAPHS * 2 * EMB;        // 2*600
  float* counts = sums + 2 * 2 * EMB;                   // 64
  float* wpack  = counts + 64;                          // up to 600*600

  auto stats_bn = [&](float* X, const float* g, const float* b, int M, int N, int relu) {
    k_zero<<<(2 * N + 255) / 256, 256, 0, stream>>>(sums, 2 * N);
    int rpb = 128;
    k_colstats<<<(M + rpb - 1) / rpb, 256, 0, stream>>>(X, sums, M, N, rpb);
    long tot = (long)M * N;
    int nb = (int)((tot + 255) / 256); if (nb > 4096) nb = 4096;
    k_bn_apply<<<nb, 256, 0, stream>>>(X, sums, g, b, M, N, relu);
  };

  // ---- conv1: t = x + scatter(x); u = t@W1+b1; BN+ReLU; h = u@W2+b2; BN0+ReLU
  hipMemcpyAsync(t, x, (size_t)N_NODES * FIRST_DIM * sizeof(float),
                 hipMemcpyDeviceToDevice, stream);
  k_scatter_edges<<<N_EDGES, 128, 0, stream>>>(x, esrc, edst, t, FIRST_DIM);
  launch_gemm<2 * EMB, FIRST_DIM>(t, c1_w1, c1_b1, u, N_NODES, wpack, stream);
  stats_bn(u, c1_g, c1_bt, N_NODES, 2 * EMB, 1);
  launch_gemm<EMB, 2 * EMB>(u, c1_w2, c1_b2, h, N_NODES, wpack, stream);
  stats_bn(h, bn0_g, bn0_b, N_NODES, EMB, 1);

  // ---- virtual node init + per-graph node counts
  k_broadcast_row<<<(N_GRAPHS * EMB + 255) / 256, 256, 0, stream>>>(vnf, ve, N_GRAPHS, EMB);
  k_zero<<<1, 64, 0, stream>>>(counts, N_GRAPHS);
  k_counts<<<(N_NODES + 255) / 256, 256, 0, stream>>>(batch, counts, N_NODES);

  // ---- 4 GIN layers with virtual node
  for (int i = 0; i < 4; ++i) {
    k_add_vnf<<<N_NODES, 128, 0, stream>>>(h, vnf, batch, EMB);
    hipMemcpyAsync(t, h, (size_t)N_NODES * EMB * sizeof(float),
                   hipMemcpyDeviceToDevice, stream);
    k_scatter_edges<<<N_EDGES, 128, 0, stream>>>(h, esrc, edst, t, EMB);
    launch_gemm<2 * EMB, EMB>(t, cw1 + (long)i * EMB * 2 * EMB, cb1 + i * 2 * EMB,
                              u, N_NODES, wpack, stream);
    stats_bn(u, cg + i * 2 * EMB, cbt + i * 2 * EMB, N_NODES, 2 * EMB, 1);
    launch_gemm<EMB, 2 * EMB>(u, cw2 + (long)i * 2 * EMB * EMB, cb2 + i * EMB,
                              h, N_NODES, wpack, stream);
    stats_bn(h, bns_g + i * EMB, bns_b + i * EMB, N_NODES, EMB, (i < 3) ? 1 : 0);

    if (i < 3) {
      // virtual-node update: pooled = segsum(h) + vnf; 2-layer MLP with BN+ReLU
      hipMemcpyAsync(pooled, vnf, (size_t)N_GRAPHS * EMB * sizeof(float),
                     hipMemcpyDeviceToDevice, stream);
      k_pool_add<<<N_NODES, 128, 0, stream>>>(h, batch, pooled, EMB);
      launch_gemm<2 * EMB, EMB>(pooled, vw1, vb1, vz, N_GRAPHS, wpack, stream);
      stats_bn(vz, vg1, vbt1, N_GRAPHS, 2 * EMB, 1);
      launch_gemm<EMB, 2 * EMB>(vz, vw2, vb2, vnf, N_GRAPHS, wpack, stream);
      stats_bn(vnf, vg2, vbt2, N_GRAPHS, EMB, 1);
    }
  }

  // ---- readout: mean pool + linear head
  k_zero<<<(N_GRAPHS * EMB + 255) / 256, 256, 0, stream>>>(pooled, (long)N_GRAPHS * EMB);
  k_pool_add<<<N_NODES, 128, 0, stream>>>(h, batch, pooled, EMB);
  k_mean_div<<<(N_GRAPHS * EMB + 255) / 256, 256, 0, stream>>>(pooled, counts, N_GRAPHS, EMB);
  launch_gemm<NUM_CLASSES, EMB>(pooled, pw, pb, out, N_GRAPHS, wpack, stream);
}